// VectorQuantizer_26113401160234
// MI455X (gfx1250) — compile-verified
//
#include <hip/hip_runtime.h>

typedef __attribute__((ext_vector_type(2))) float v2f;
typedef __attribute__((ext_vector_type(4))) float v4f;
typedef __attribute__((ext_vector_type(8))) float v8f;

#define N_EMB   4096
#define EMB_DIM 256
#define N_ROWS  32768          // 32*32*32
#define ZQ_ELEMS 8388608       // 32*256*32*32
#define ROWPAD  260            // 256 + 4 pad -> conflict-free banks, 16B aligned rows

// ---------------------------------------------------------------- init hist
__global__ void vq_init_kernel(unsigned* __restrict__ hist) {
    int t = blockIdx.x * blockDim.x + threadIdx.x;
    if (t < N_EMB) hist[t] = 0u;
}

// ---------------------------------------------------- ||e||^2 per codebook row
__global__ __launch_bounds__(256) void vq_e2_kernel(const float* __restrict__ emb,
                                                    float* __restrict__ e2) {
    int wave = (blockIdx.x * blockDim.x + threadIdx.x) >> 5;
    int lane = threadIdx.x & 31;
    if (wave >= N_EMB) return;
    const float* row = emb + (size_t)wave * EMB_DIM;
    float s = 0.f;
#pragma unroll
    for (int i = 0; i < EMB_DIM; i += 32) {
        float v = row[i + lane];
        s += v * v;
    }
#pragma unroll
    for (int m = 16; m >= 1; m >>= 1) s += __shfl_xor(s, m, 32);
    if (lane == 0) e2[wave] = s;
}

// ------------------------------------------- fused WMMA GEMM + argmin kernel
// 8 waves/WG; each wave owns a 16-row tile (A register-resident), all waves
// share a 32-code B tile staged in LDS (8x less L2 traffic than per-wave
// streaming). Register double-buffering overlaps next tile's global loads
// with the 128-WMMA compute phase. score = ||e||^2 - 2 z.e (||z||^2 const).
__global__ __launch_bounds__(256) void vq_argmin_kernel(
        const float* __restrict__ z,      // (32,256,32,32)
        const float* __restrict__ emb,    // (4096,256)
        const float* __restrict__ e2,     // (4096)
        int* __restrict__ idx_out,        // (32768)
        unsigned* __restrict__ hist) {    // (4096)
    __shared__ float lbuf[32 * ROWPAD];   // 33,280 B

    int tid  = threadIdx.x;
    int lane = tid & 31;
    int wave = tid >> 5;
    int tile = blockIdx.x * 8 + wave;     // 2048 tiles of 16 rows
    int rowbase = tile * 16;
    int l16  = lane & 15;
    int half = lane >> 4;

    // A addressing: zf[row][c] = z[b][c][h][w], row = b*1024 + h*32 + w.
    // A is stage-invariant -> compiler keeps the 64 v2f fragments in VGPRs.
    int bb = rowbase >> 10;
    int hw = (rowbase & 1023) + l16;
    const float* abase = z + ((size_t)bb << 18) + hw;   // + (c<<10) selects channel

    float bestd[8];
    int   besti[8];
#pragma unroll
    for (int j = 0; j < 8; ++j) { bestd[j] = 3.402823466e38f; besti[j] = 0; }

    // prefetch stage 0 B tile into registers (32 rows x 256 f32, coalesced v4f)
    v4f pf[8];
#pragma unroll
    for (int i = 0; i < 8; ++i) {
        int lin = tid + 256 * i;
        int row = lin >> 6;
        int c4  = (lin & 63) << 2;
        pf[i] = *(const v4f*)(emb + (size_t)row * EMB_DIM + c4);
    }

    for (int s = 0; s < N_EMB / 32; ++s) {
        // commit prefetched tile into LDS
#pragma unroll
        for (int i = 0; i < 8; ++i) {
            int lin = tid + 256 * i;
            int row = lin >> 6;
            int c4  = (lin & 63) << 2;
            *(v4f*)(&lbuf[row * ROWPAD + c4]) = pf[i];
        }
        __syncthreads();

        // issue next stage's global loads; latency hidden by compute below
        if (s + 1 < N_EMB / 32) {
            const float* src = emb + (size_t)(s + 1) * 32 * EMB_DIM;
#pragma unroll
            for (int i = 0; i < 8; ++i) {
                int lin = tid + 256 * i;
                int row = lin >> 6;
                int c4  = (lin & 63) << 2;
                pf[i] = *(const v4f*)(src + (size_t)row * EMB_DIM + c4);
            }
        }

        int codebase = s * 32;
        // B fragment from LDS: lane holds B[K=2*half+{0,1}][N=l16]
        const float* bb0 = &lbuf[l16 * ROWPAD + 2 * half];
        const float* bb1 = bb0 + 16 * ROWPAD;
        v8f acc0 = {};
        v8f acc1 = {};
#pragma unroll
        for (int kk = 0; kk < EMB_DIM / 4; ++kk) {
            int c0 = 4 * kk + 2 * half;
            v2f a;
            a.x = abase[(size_t)c0 << 10];
            a.y = abase[(size_t)(c0 + 1) << 10];
            v2f b0 = *(const v2f*)(bb0 + 4 * kk);
            v2f b1 = *(const v2f*)(bb1 + 4 * kk);
            acc0 = __builtin_amdgcn_wmma_f32_16x16x4_f32(
                    false, a, false, b0, (short)0, acc0, false, false);
            acc1 = __builtin_amdgcn_wmma_f32_16x16x4_f32(
                    false, a, false, b1, (short)0, acc1, false, false);
        }

        // chain 0 first (lower code indices win ties via strict '<')
        float e2v0 = e2[codebase + l16];
        float e2v1 = e2[codebase + 16 + l16];
        int code0 = codebase + l16;
        int code1 = codebase + 16 + l16;
#pragma unroll
        for (int j = 0; j < 8; ++j) {
            float s0 = e2v0 - 2.0f * acc0[j];
            if (s0 < bestd[j]) { bestd[j] = s0; besti[j] = code0; }
            float s1 = e2v1 - 2.0f * acc1[j];
            if (s1 < bestd[j]) { bestd[j] = s1; besti[j] = code1; }
        }
        __syncthreads();   // all waves done reading lbuf before next commit
    }

    // reduce across the 16 lanes holding the same rows
#pragma unroll
    for (int m = 8; m >= 1; m >>= 1) {
#pragma unroll
        for (int j = 0; j < 8; ++j) {
            float od = __shfl_xor(bestd[j], m, 32);
            int   oi = __shfl_xor(besti[j], m, 32);
            if (od < bestd[j] || (od == bestd[j] && oi < besti[j])) {
                bestd[j] = od; besti[j] = oi;
            }
        }
    }
    if (l16 == 0) {
#pragma unroll
        for (int j = 0; j < 8; ++j) {
            int row = rowbase + j + 8 * half;   // lane0 -> rows 0..7, lane16 -> 8..15
            idx_out[row] = besti[j];
            atomicAdd(&hist[besti[j]], 1u);
        }
    }
}

// ----------------------------------- gather z_q, write (B,C,H,W), MSE partials
__global__ __launch_bounds__(256) void vq_quantize_kernel(
        const float* __restrict__ z, const float* __restrict__ emb,
        const int* __restrict__ idx, float* __restrict__ zq_out,
        float* __restrict__ idxf_out, float* __restrict__ blocksum) {
    __shared__ float sred[256];
    int t = blockIdx.x * 256 + threadIdx.x;   // flat (B,C,H,W) index
    int w = t & 31, h = (t >> 5) & 31, c = (t >> 10) & 255, b = t >> 18;
    int r = (b << 10) + (h << 5) + w;         // row in zf view
    int ix = idx[r];
    float zq = emb[(size_t)ix * EMB_DIM + c];
    float zp = z[t];                          // z is already (B,C,H,W)
    zq_out[t] = zq;                           // z_q_st == z_q numerically
    if (c == 0) idxf_out[r] = (float)ix;
    float d = zq - zp;
    sred[threadIdx.x] = d * d;
    __syncthreads();
    for (int s = 128; s > 0; s >>= 1) {
        if (threadIdx.x < s) sred[threadIdx.x] += sred[threadIdx.x + s];
        __syncthreads();
    }
    if (threadIdx.x == 0) blocksum[blockIdx.x] = sred[0];
}

// ---------------------------------------------------- one-hot encodings (512MB)
__global__ __launch_bounds__(256) void vq_onehot_kernel(const int* __restrict__ idx,
                                                        float* __restrict__ out) {
    size_t t = (size_t)blockIdx.x * 256 + threadIdx.x;  // 33,554,432 threads
    int row = (int)(t >> 10);
    int col = (int)(t & 1023) << 2;
    int ix = idx[row];
    v4f v;
    v.x = (ix == col + 0) ? 1.0f : 0.0f;
    v.y = (ix == col + 1) ? 1.0f : 0.0f;
    v.z = (ix == col + 2) ? 1.0f : 0.0f;
    v.w = (ix == col + 3) ? 1.0f : 0.0f;
    *(v4f*)(out + (t << 2)) = v;
}

// ------------------------------------------- scalars: loss + perplexity
__global__ __launch_bounds__(1024) void vq_finalize_kernel(
        const unsigned* __restrict__ hist, const float* __restrict__ blocksum,
        float* __restrict__ loss_out, float* __restrict__ perp_out) {
    __shared__ float sh[1024];
    __shared__ float sl[1024];
    int t = threadIdx.x;
    float hsum = 0.f;
    for (int i = t; i < N_EMB; i += 1024) {
        float p = (float)hist[i] * (1.0f / (float)N_ROWS);
        hsum += -p * logf(p + 1e-10f);
    }
    float lsum = 0.f;
    for (int i = t; i < 32768; i += 1024) lsum += blocksum[i];
    sh[t] = hsum; sl[t] = lsum;
    __syncthreads();
    for (int s = 512; s > 0; s >>= 1) {
        if (t < s) { sh[t] += sh[t + s]; sl[t] += sl[t + s]; }
        __syncthreads();
    }
    if (t == 0) {
        *perp_out = expf(sh[0]);
        *loss_out = 1.25f * sl[0] / (float)ZQ_ELEMS;   // q_loss + BETA*e_loss
    }
}

extern "C" void kernel_launch(void* const* d_in, const int* in_sizes, int n_in,
                              void* d_out, int out_size, void* d_ws, size_t ws_size,
                              hipStream_t stream) {
    const float* z   = (const float*)d_in[0];   // (32,256,32,32)
    const float* emb = (const float*)d_in[1];   // (4096,256)
    float* out = (float*)d_out;

    // workspace layout
    float*    e2   = (float*)d_ws;              // 4096 f32
    unsigned* hist = (unsigned*)(e2 + N_EMB);   // 4096 u32
    int*      idx  = (int*)(hist + N_EMB);      // 32768 i32
    float*    bsum = (float*)(idx + N_ROWS);    // 32768 f32

    // output layout: loss | z_q(8388608) | perplexity | min_encodings(32768*4096) | idx(32768)
    float* loss_out = out;
    float* zq_out   = out + 1;
    float* perp_out = out + 1 + (size_t)ZQ_ELEMS;
    float* onehot   = out + 2 + (size_t)ZQ_ELEMS;
    float* idxf     = onehot + (size_t)N_ROWS * N_EMB;

    hipLaunchKernelGGL(vq_init_kernel, dim3(16), dim3(256), 0, stream, hist);
    hipLaunchKernelGGL(vq_e2_kernel, dim3(512), dim3(256), 0, stream, emb, e2);
    hipLaunchKernelGGL(vq_argmin_kernel, dim3(256), dim3(256), 0, stream,
                       z, emb, e2, idx, hist);
    hipLaunchKernelGGL(vq_quantize_kernel, dim3(32768), dim3(256), 0, stream,
                       z, emb, idx, zq_out, idxf, bsum);
    hipLaunchKernelGGL(vq_onehot_kernel, dim3(131072), dim3(256), 0, stream,
                       idx, onehot);
    hipLaunchKernelGGL(vq_finalize_kernel, dim3(1), dim3(1024), 0, stream,
                       hist, bsum, loss_out, perp_out);
}